// LQR_45002667327505
// MI455X (gfx1250) — compile-verified
//
#include <hip/hip_runtime.h>
#include <hip/hip_bf16.h>

// ---------------------------------------------------------------------------
// LQR (Riccati backward scan + forward rollout) for MI455X / gfx1250.
// NB=128 T=128 NS=64 NC=32 N=96, all float32.
//
// v2 changes vs v1 (driven by asm inspection):
//  * V, F, K held in LDS in K-pair-interleaved layout so every WMMA A/B
//    fragment is a single aligned ds_load_b64 (was 2x strided ds_load_b32).
//  * Per-tile base pointers + unrolled constant K offsets -> fragment
//    addresses fold into ds_load immediate offsets (kills v_add chains).
//  * Two WMMA accumulators per tile to break the D->C RAW chain.
// ---------------------------------------------------------------------------

#define NB 128
#define TT 128
#define NS 64
#define NC 32
#define NN 96   // NS + NC

typedef __attribute__((ext_vector_type(2))) float v2f;
typedef __attribute__((ext_vector_type(8))) float v8f;

// LDS layout (floats)
#define OFF_V    0                      // Vp: 64*64, pair-interleaved over k
#define OFF_F    (OFF_V   + NS*NS)      // Fp: 64*96, pair-interleaved (reused: Kp 32*64)
#define OFF_FTV  (OFF_F   + NS*NN)      // FtV 96*64 row-major (reused: Aug 32x97, P 64x64)
#define OFF_Q    (OFF_FTV + NN*NS)      // Qbar 96*96 row-major
#define OFF_QV   (OFF_Q   + NN*NN)      // 96  qbar
#define OFF_SV   (OFF_QV  + NN)         // 64  v vector
#define OFF_C    (OFF_SV  + NS)         // 64  c vector
#define OFF_M    (OFF_C   + NS)         // 32  GJ multipliers
#define OFF_KV   (OFF_M   + NC)         // 32  kvec
#define SMEM_FLOATS (OFF_KV + NC)

// pair-interleaved index: X[k][n] (row stride W) -> Xp[((k>>1)*W + n)*2 + (k&1)]
__device__ __forceinline__ int pidx(int k, int n, int W) {
    return (((k >> 1) * W + n) << 1) + (k & 1);
}

__global__ __launch_bounds__(256)
void lqr_backward_kernel(const float* __restrict__ Q, const float* __restrict__ p,
                         const float* __restrict__ A, const float* __restrict__ B,
                         const float* __restrict__ c1,
                         float* __restrict__ Kall, float* __restrict__ kall)
{
    extern __shared__ float smem[];
    float* sVp  = smem + OFF_V;    // V pair-interleaved
    float* sFp  = smem + OFF_F;    // F pair-interleaved (later: K pair-interleaved)
    float* sFtV = smem + OFF_FTV;  // FtV row-major     (later: Aug, then P)
    float* sQ   = smem + OFF_Q;    // Qbar row-major
    float* sq   = smem + OFF_QV;
    float* sv   = smem + OFF_SV;
    float* sc   = smem + OFF_C;
    float* sm   = smem + OFF_M;
    float* skv  = smem + OFF_KV;

    const int b    = blockIdx.x;
    const int tid  = threadIdx.x;
    const int wave = tid >> 5;
    const int lane = tid & 31;
    const int lo   = lane & 15;   // 0..15
    const int hi   = lane >> 4;   // 0 or 1

    for (int i = tid; i < NS*NS; i += 256) sVp[i] = 0.0f;
    if (tid < NS) sv[tid] = 0.0f;
    __syncthreads();

    for (int t = TT - 1; t >= 0; --t) {
        // ---- stage inputs into LDS -------------------------------------
        const float* Qbt = Q  + ((size_t)b*TT + t) * (NN*NN);
        const float* pbt = p  + ((size_t)b*TT + t) * NN;
        const float* Abt = A  + ((size_t)b*TT + t) * (NS*NS);
        const float* Bbt = B  + ((size_t)b*TT + t) * (NS*NC);
        const float* cbt = c1 + ((size_t)b*TT + t) * NS;

        for (int i = tid; i < NN*NN; i += 256) sQ[i] = Qbt[i];
        for (int i = tid; i < NS*NN; i += 256) {
            int r = i / NN, cc = i % NN;
            float val = (cc < NS) ? Abt[r*NS + cc] : Bbt[r*NC + (cc - NS)];
            sFp[pidx(r, cc, NN)] = val;
        }
        if (tid < NN) sq[tid] = pbt[tid];
        if (tid < NS) sc[tid] = cbt[tid];

        // prefetch next step's Q into cache while we crunch this step
        if (t > 0) {
            const float* Qn = Q + ((size_t)b*TT + (t-1)) * (NN*NN);
            if (tid < (NN*NN*4)/256)
                __builtin_prefetch(Qn + (size_t)tid*64, 0, 1);
        }
        __syncthreads();

        // ---- FtV = F^T * V   (96x64), 24 tiles over 8 waves ------------
        for (int tile = wave; tile < 24; tile += 8) {
            const int I0 = (tile >> 2) * 16, J0 = (tile & 3) * 16;
            // A[m][k] = F[k][I0+m]: pair base at k-pair 'hi', col I0+lo
            const v2f* aB = (const v2f*)&sFp[(hi*NN + I0 + lo) << 1];
            const v2f* bB = (const v2f*)&sVp[(hi*NS + J0 + lo) << 1];
            v8f acc0 = {}, acc1 = {};
#pragma unroll
            for (int kb = 0; kb < NS; kb += 8) {
                v2f a0 = aB[((kb    ) >> 1) * NN];
                v2f b0 = bB[((kb    ) >> 1) * NS];
                v2f a1 = aB[((kb + 4) >> 1) * NN];
                v2f b1 = bB[((kb + 4) >> 1) * NS];
                acc0 = __builtin_amdgcn_wmma_f32_16x16x4_f32(
                           false, a0, false, b0, (short)0, acc0, false, false);
                acc1 = __builtin_amdgcn_wmma_f32_16x16x4_f32(
                           false, a1, false, b1, (short)0, acc1, false, false);
            }
#pragma unroll
            for (int r = 0; r < 8; ++r)
                sFtV[(I0 + r + 8*hi)*NS + J0 + lo] = acc0[r] + acc1[r];
        }
        __syncthreads();

        // ---- Qbar += FtV * F  (96x96), 36 tiles over 8 waves -----------
        for (int tile = wave; tile < 36; tile += 8) {
            const int I0 = (tile / 6) * 16, J0 = (tile % 6) * 16;
            const v2f* aB = (const v2f*)&sFtV[(I0 + lo)*NS + 2*hi]; // row-contig pair
            const v2f* bB = (const v2f*)&sFp[(hi*NN + J0 + lo) << 1];
            v8f acc0 = {}, acc1 = {};
#pragma unroll
            for (int kb = 0; kb < NS; kb += 8) {
                v2f a0 = aB[ kb      >> 1];
                v2f b0 = bB[( kb      >> 1) * NN];
                v2f a1 = aB[(kb + 4) >> 1];
                v2f b1 = bB[((kb + 4) >> 1) * NN];
                acc0 = __builtin_amdgcn_wmma_f32_16x16x4_f32(
                           false, a0, false, b0, (short)0, acc0, false, false);
                acc1 = __builtin_amdgcn_wmma_f32_16x16x4_f32(
                           false, a1, false, b1, (short)0, acc1, false, false);
            }
#pragma unroll
            for (int r = 0; r < 8; ++r)
                sQ[(I0 + r + 8*hi)*NN + J0 + lo] += acc0[r] + acc1[r];
        }

        // ---- qbar = p + FtV*c + F^T*v  (96 matvec, VALU) ---------------
        if (tid < NN) {
            float acc = sq[tid];
            for (int k = 0; k < NS; ++k) acc += sFtV[tid*NS + k] * sc[k];
            for (int j = 0; j < NS; ++j) acc += sFp[pidx(j, tid, NN)] * sv[j];
            sq[tid] = acc;
        }
        __syncthreads();

        // ---- Gauss-Jordan: [Quu | Qux | qu] -> [I | Quu^-1 Qux | Quu^-1 qu]
        float* sAug = sFtV;  // 32 x 97 (FtV consumed)
        for (int i = tid; i < NC*97; i += 256) {
            int r = i / 97, cc = i % 97;
            float val;
            if (cc < NC)       val = sQ[(NS + r)*NN + NS + cc];   // Quu
            else if (cc < 96)  val = sQ[(NS + r)*NN + (cc - NC)]; // Qux
            else               val = sq[NS + r];                  // qu
            sAug[i] = val;
        }
        __syncthreads();

        for (int k = 0; k < NC; ++k) {
            const float piv = sAug[k*97 + k];
            __syncthreads();
            const float inv = 1.0f / piv;
            if (tid < 97) sAug[k*97 + tid] *= inv;
            if (tid < NC) sm[tid] = sAug[tid*97 + k];
            __syncthreads();
            for (int i = tid; i < NC*97; i += 256) {
                int r = i / 97, cc = i % 97;
                if (r != k) sAug[i] -= sm[r] * sAug[k*97 + cc];
            }
            __syncthreads();
        }

        // ---- K = -sol[:,:64] (pair-interleaved in LDS), kvec = -sol[:,96]
        float* sKp = sFp;  // reuse F space (F consumed); K is 32x64
        float* Kbt = Kall + ((size_t)b*TT + t) * (NC*NS);
        for (int i = tid; i < NC*NS; i += 256) {
            int r = i / NS, cc = i % NS;
            float kv = -sAug[r*97 + NC + cc];
            sKp[pidx(r, cc, NS)] = kv;
            Kbt[i] = kv;
        }
        if (tid < NC) {
            float kv = -sAug[tid*97 + 96];
            skv[tid] = kv;
            kall[((size_t)b*TT + t)*NC + tid] = kv;
        }
        __syncthreads();

        // ---- P = Qxu * K  (64x64), 16 tiles over 8 waves ---------------
        float* sP = sFtV;  // reuse (Aug consumed)
        for (int tile = wave; tile < 16; tile += 8) {
            const int I0 = (tile >> 2) * 16, J0 = (tile & 3) * 16;
            const v2f* aB = (const v2f*)&sQ[(I0 + lo)*NN + NS + 2*hi]; // Qxu row pair
            const v2f* bB = (const v2f*)&sKp[(hi*NS + J0 + lo) << 1];
            v8f acc0 = {}, acc1 = {};
#pragma unroll
            for (int kb = 0; kb < NC; kb += 8) {
                v2f a0 = aB[ kb      >> 1];
                v2f b0 = bB[( kb      >> 1) * NS];
                v2f a1 = aB[(kb + 4) >> 1];
                v2f b1 = bB[((kb + 4) >> 1) * NS];
                acc0 = __builtin_amdgcn_wmma_f32_16x16x4_f32(
                           false, a0, false, b0, (short)0, acc0, false, false);
                acc1 = __builtin_amdgcn_wmma_f32_16x16x4_f32(
                           false, a1, false, b1, (short)0, acc1, false, false);
            }
#pragma unroll
            for (int r = 0; r < 8; ++r)
                sP[(I0 + r + 8*hi)*NS + J0 + lo] = acc0[r] + acc1[r];
        }
        __syncthreads();

        // ---- V = 0.5*((Qxx+P) + (Qxx+P)^T) (pair layout);  v = qx + Qxu*kvec
        for (int i = tid; i < NS*NS; i += 256) {
            int r = i / NS, cc = i % NS;
            float vij = sQ[r*NN + cc] + sP[r*NS + cc];
            float vji = sQ[cc*NN + r] + sP[cc*NS + r];
            sVp[pidx(r, cc, NS)] = 0.5f * (vij + vji);
        }
        if (tid < NS) {
            float acc = sq[tid];
            for (int j = 0; j < NC; ++j) acc += sQ[tid*NN + NS + j] * skv[j];
            sv[tid] = acc;
        }
        __syncthreads();
    }
}

__global__ __launch_bounds__(128)
void lqr_forward_kernel(const float* __restrict__ Kall, const float* __restrict__ kall,
                        const float* __restrict__ A, const float* __restrict__ B,
                        const float* __restrict__ c1, const float* __restrict__ x_init,
                        float* __restrict__ tau)
{
    __shared__ float sx[NS];
    __shared__ float su[NC];
    const int b   = blockIdx.x;
    const int tid = threadIdx.x;

    if (tid < NS) sx[tid] = x_init[(size_t)b*NS + tid];
    __syncthreads();

    for (int t = 0; t < TT; ++t) {
        const float* Kbt = Kall + ((size_t)b*TT + t) * (NC*NS);
        const float* kbt = kall + ((size_t)b*TT + t) * NC;
        const float* Abt = A    + ((size_t)b*TT + t) * (NS*NS);
        const float* Bbt = B    + ((size_t)b*TT + t) * (NS*NC);
        const float* cbt = c1   + ((size_t)b*TT + t) * NS;
        float* taubt     = tau  + ((size_t)b*TT + t) * NN;

        if (tid < NC) {                    // u = K x + k
            float acc = kbt[tid];
            for (int j = 0; j < NS; ++j) acc += Kbt[tid*NS + j] * sx[j];
            su[tid] = acc;
        }
        __syncthreads();

        if (tid < NS)            taubt[tid] = sx[tid];      // tau = [x, u]
        else if (tid < NN)       taubt[tid] = su[tid - NS];

        float xn = 0.0f;
        if (tid < NS) {                    // xn = A x + B u + c
            xn = cbt[tid];
            for (int j = 0; j < NS; ++j) xn += Abt[tid*NS + j] * sx[j];
            for (int j = 0; j < NC; ++j) xn += Bbt[tid*NC + j] * su[j];
        }
        __syncthreads();
        if (tid < NS) sx[tid] = xn;
        __syncthreads();
    }
}

extern "C" void kernel_launch(void* const* d_in, const int* in_sizes, int n_in,
                              void* d_out, int out_size, void* d_ws, size_t ws_size,
                              hipStream_t stream) {
    const float* Q      = (const float*)d_in[0];
    const float* p      = (const float*)d_in[1];
    const float* A      = (const float*)d_in[2];
    const float* B      = (const float*)d_in[3];
    const float* c1     = (const float*)d_in[4];
    const float* x_init = (const float*)d_in[5];
    float* tau = (float*)d_out;

    float* Kall = (float*)d_ws;                       // NB*T*NC*NS
    float* kall = Kall + (size_t)NB*TT*NC*NS;         // NB*T*NC

    const size_t smemBytes = (size_t)SMEM_FLOATS * sizeof(float);  // ~101 KB
    hipFuncSetAttribute((const void*)lqr_backward_kernel,
                        hipFuncAttributeMaxDynamicSharedMemorySize, (int)smemBytes);

    lqr_backward_kernel<<<NB, 256, smemBytes, stream>>>(Q, p, A, B, c1, Kall, kall);
    lqr_forward_kernel<<<NB, 128, 0, stream>>>(Kall, kall, A, B, c1, x_init, tau);
}